// SoftHungarian_90855738179688
// MI455X (gfx1250) — compile-verified
//
#include <hip/hip_runtime.h>
#include <hip/hip_bf16.h>

typedef __attribute__((ext_vector_type(2))) float v2f;
typedef __attribute__((ext_vector_type(8))) float v8f;

#define BB   32
#define HN   320
#define HD   512
#define FINF 3.4e38f

// ---------------------------------------------------------------------------
// Kernel 1: per-row inverse norms.  One wave32 per row of 512 floats.
// inv[row] = 1 / max(||x||, 1e-12)
// ---------------------------------------------------------------------------
__global__ __launch_bounds__(256) void norm_kernel(const float* __restrict__ t1,
                                                   const float* __restrict__ t2,
                                                   float* __restrict__ inv1,
                                                   float* __restrict__ inv2) {
    const int w    = blockIdx.x * (blockDim.x >> 5) + (threadIdx.x >> 5);
    const int lane = threadIdx.x & 31;
    const int rows = BB * HN;
    if (w >= 2 * rows) return;
    const float* src;
    float* dst;
    int row;
    if (w < rows) { src = t1; dst = inv1; row = w; }
    else          { src = t2; dst = inv2; row = w - rows; }
    const float* p = src + (size_t)row * HD;
    float s = 0.0f;
    #pragma unroll
    for (int k = lane; k < HD; k += 32) { float x = p[k]; s += x * x; }
    #pragma unroll
    for (int off = 16; off > 0; off >>= 1) s += __shfl_down(s, off, 32);
    if (lane == 0) dst[row] = 1.0f / fmaxf(sqrtf(s), 1e-12f);
}

// ---------------------------------------------------------------------------
// Kernel 2: cost[b,n,m] = 1 - |dot(t1[b,n],t2[b,m]) * inv1[b,n] * inv2[b,m]|
// Batched 320x320x512 fp32 GEMM via V_WMMA_F32_16X16X4_F32.
// One wave per 16x16 output tile; 8 waves (2x4 tiles) per block.
// A 16x4 layout: lanes 0-15 -> M=lane, K=0,1 ; lanes 16-31 -> M=lane-16, K=2,3
// B 4x16 layout mirrors A (row striped across lanes, halves offset by 2 in K).
// C 16x16 f32: VGPR r -> row (r + 8*half), col = lane&15.
// ---------------------------------------------------------------------------
__global__ __launch_bounds__(256) void cost_wmma_kernel(const float* __restrict__ t1,
                                                        const float* __restrict__ t2,
                                                        const float* __restrict__ inv1,
                                                        const float* __restrict__ inv2,
                                                        float* __restrict__ cost) {
    const int b    = blockIdx.z;
    const int wv   = threadIdx.x >> 5;          // 0..7
    const int lane = threadIdx.x & 31;
    const int half = lane >> 4;                 // 0/1
    const int mrow = lane & 15;                 // 0..15
    const int m0   = (blockIdx.x * 4 + (wv & 3)) * 16;   // t2 row / cost col tile
    const int n0   = (blockIdx.y * 2 + (wv >> 2)) * 16;  // t1 row / cost row tile

    const float* arow = t1 + ((size_t)b * HN + n0 + mrow) * HD + 2 * half;
    const float* brow = t2 + ((size_t)b * HN + m0 + mrow) * HD + 2 * half;

    v8f c = {};
    #pragma unroll 4
    for (int kk = 0; kk < HD; kk += 4) {
        v2f a = *(const v2f*)(arow + kk);   // K = kk+2*half, kk+2*half+1
        v2f bfrag = *(const v2f*)(brow + kk);
        c = __builtin_amdgcn_wmma_f32_16x16x4_f32(false, a, false, bfrag,
                                                  (short)0, c, false, false);
    }

    const float im = inv2[b * HN + m0 + mrow];
    #pragma unroll
    for (int r = 0; r < 8; ++r) {
        const int n  = n0 + r + 8 * half;
        const float d = 1.0f - fabsf(c[r] * inv1[b * HN + n] * im);
        cost[((size_t)b * HN + n) * HN + (m0 + mrow)] = d;
    }
}

// ---------------------------------------------------------------------------
// Kernel 3: Jonker-Volgenant shortest-augmenting-path assignment, one block
// per batch, 320 threads (thread t owns column j = t+1).  All solver state in
// LDS; column-parallel relax + tree argmin each step.  Finishes with the
// matched-cost mean.
// ---------------------------------------------------------------------------
__global__ __launch_bounds__(HN) void hungarian_kernel(const float* __restrict__ cost,
                                                       float* __restrict__ out) {
    const int b   = blockIdx.x;
    const float* C = cost + (size_t)b * HN * HN;
    const int tid = threadIdx.x;
    const int j   = tid + 1;                    // column 1..320

    __shared__ float u[HN + 1], v[HN + 1], minv[HN + 1];
    __shared__ int   p[HN + 1], way[HN + 1];
    __shared__ unsigned char used[HN + 1];
    __shared__ float rv[512];
    __shared__ int   ri[512];
    __shared__ int   s_j0, s_j1, s_i0;
    __shared__ float s_delta;

    u[j] = 0.0f; v[j] = 0.0f; p[j] = 0; way[j] = 0;
    if (tid == 0) { u[0] = 0.0f; v[0] = 0.0f; p[0] = 0; way[0] = 0; }
    if (tid < 512 - HN) { rv[tid + HN] = FINF; ri[tid + HN] = 1; }  // argmin pad
    __syncthreads();

    for (int i = 1; i <= HN; ++i) {
        if (tid == 0) { p[0] = i; s_j0 = 0; }
        minv[j] = FINF;
        used[j] = 0;
        if (tid == 0) used[0] = 0;
        __syncthreads();

        while (true) {
            if (tid == 0) { used[s_j0] = 1; s_i0 = p[s_j0]; }
            __syncthreads();
            const int i0 = s_i0;
            const int jc = s_j0;
            const float* crow = C + (size_t)(i0 - 1) * HN;

            const bool freej = (used[j] == 0);
            float mvj = minv[j];
            if (freej) {
                const float cur = crow[j - 1] - u[i0] - v[j];
                if (cur < mvj) { mvj = cur; minv[j] = cur; way[j] = jc; }
            }
            rv[tid] = freej ? mvj : FINF;
            ri[tid] = j;
            __syncthreads();
            #pragma unroll
            for (int s = 256; s > 0; s >>= 1) {
                if (tid < s) {
                    const float o = rv[tid + s];
                    const int   oi = ri[tid + s];
                    if (o < rv[tid] || (o == rv[tid] && oi < ri[tid])) {
                        rv[tid] = o; ri[tid] = oi;
                    }
                }
                __syncthreads();
            }
            if (tid == 0) { s_delta = rv[0]; s_j1 = ri[0]; }
            __syncthreads();

            const float delta = s_delta;
            if (used[j]) { u[p[j]] += delta; v[j] -= delta; }
            else         { minv[j] -= delta; }
            if (tid == 0) { u[p[0]] += delta; v[0] -= delta; }  // column 0 is used
            __syncthreads();
            if (tid == 0) s_j0 = s_j1;
            __syncthreads();
            if (p[s_j0] == 0) break;
        }

        if (tid == 0) {            // augment along alternating tree
            int j0 = s_j0;
            while (j0 != 0) { const int j1 = way[j0]; p[j0] = p[j1]; j0 = j1; }
        }
        __syncthreads();
    }

    // matched mean: column j matched to row p[j]-1
    rv[tid] = C[(size_t)(p[j] - 1) * HN + (j - 1)];
    if (tid < 512 - HN) rv[tid + HN] = 0.0f;
    __syncthreads();
    #pragma unroll
    for (int s = 256; s > 0; s >>= 1) {
        if (tid < s) rv[tid] += rv[tid + s];
        __syncthreads();
    }
    if (tid == 0) out[b] = rv[0] / (float)HN;
}

// ---------------------------------------------------------------------------
// Launch.  Inputs: 0=dense_t1, 1=dense_t2 (the GAT weights are dead code).
// Workspace: inv1 (B*N) | inv2 (B*M) | cost (B*N*M)  ~= 13.2 MB.
// ---------------------------------------------------------------------------
extern "C" void kernel_launch(void* const* d_in, const int* in_sizes, int n_in,
                              void* d_out, int out_size, void* d_ws, size_t ws_size,
                              hipStream_t stream) {
    const float* t1 = (const float*)d_in[0];
    const float* t2 = (const float*)d_in[1];

    float* inv1 = (float*)d_ws;
    float* inv2 = inv1 + BB * HN;
    float* cost = inv2 + BB * HN;

    // 2 * B * 320 rows, one wave each, 8 waves per block
    const int total_waves = 2 * BB * HN;
    norm_kernel<<<total_waves / 8, 256, 0, stream>>>(t1, t2, inv1, inv2);

    // 20x20 tiles of 16x16 per batch; 8 tiles (2 rows x 4 cols) per block
    dim3 g2(HN / 16 / 4, HN / 16 / 2, BB);  // (5, 10, 32)
    cost_wmma_kernel<<<g2, 256, 0, stream>>>(t1, t2, inv1, inv2, cost);

    hungarian_kernel<<<BB, HN, 0, stream>>>(cost, (float*)d_out);
}